// AttentionEdgeDecoder_62122406969713
// MI455X (gfx1250) — compile-verified
//
#include <hip/hip_runtime.h>
#include <hip/hip_bf16.h>
#include <math.h>

typedef __attribute__((ext_vector_type(2))) float v2f;
typedef __attribute__((ext_vector_type(4))) float v4f;
typedef __attribute__((ext_vector_type(8))) float v8f;

#define HID 128
#define NH 8
#define NROWS 4096
#define SEQ 4097

// ---------------------------------------------------------------------------
// k1: q[b] = hg[b] @ W_q ; wfold[b,h,d] = 0.25 * sum_t W_kv[d, h*16+t] * q[b,h*16+t]
// wfold padded to 16 "heads" (rows 8..15 zero) so the WMMA A-matrix loads are
// unconditional for all 32 lanes.
// ---------------------------------------------------------------------------
__global__ void k1_fold(const float* __restrict__ h, const float* __restrict__ Wq,
                        const float* __restrict__ Wkv, float* __restrict__ wfold) {
    int b = blockIdx.x;          // 4 blocks
    int t = threadIdx.x;         // 128 threads
    __shared__ float hg[HID];
    __shared__ float q[HID];
    hg[t] = h[((size_t)b * SEQ + NROWS) * HID + t];
    __syncthreads();
    float acc = 0.f;
    #pragma unroll 4
    for (int d = 0; d < HID; ++d) acc += hg[d] * Wq[d * HID + t];
    q[t] = acc;
    __syncthreads();
    for (int hh = 0; hh < 16; ++hh) {
        float a = 0.f;
        if (hh < NH) {
            #pragma unroll
            for (int u = 0; u < 16; ++u) a += Wkv[t * 256 + hh * 16 + u] * q[hh * 16 + u];
            a *= 0.25f;          // 1/sqrt(D), D=16
        }
        wfold[(b * 16 + hh) * HID + t] = a;
    }
}

// ---------------------------------------------------------------------------
// k2: scores s[b,h,n] = hn[b,n,:] . wfold[b,h,:]  via V_WMMA_F32_16X16X4_F32
// One wave per 16-row n-tile. A = wfold (M=16 heads x K=4), B = hn columns
// (K=4 x N=16 rows), C accumulates 16x16 over 32 K-steps (hid=128).
// ISA layout: lane<16 holds K={0,1}, lane>=16 holds K={2,3} -> float2 load at
// row_base + k0 + 2*(lane>>4) for BOTH A and B.
// ---------------------------------------------------------------------------
__global__ void k2_scores(const float* __restrict__ h, const float* __restrict__ wfold,
                          float* __restrict__ sbuf) {
    int wid  = blockIdx.x;        // 4 * 256 waves
    int b    = wid >> 8;
    int n0   = (wid & 255) * 16;
    int lane = threadIdx.x;       // 32 (one wave)
    int lo   = lane & 15;
    int hi   = lane >> 4;

    const float* arow = wfold + (size_t)(b * 16 + lo) * HID + 2 * hi;            // head row
    const float* brow = h + ((size_t)b * SEQ + n0 + lo) * HID + 2 * hi;          // hn row

    v8f c = {};
    #pragma unroll 4
    for (int k0 = 0; k0 < HID; k0 += 4) {
        v2f a  = *(const v2f*)(arow + k0);
        v2f bb = *(const v2f*)(brow + k0);
        c = __builtin_amdgcn_wmma_f32_16x16x4_f32(false, a, false, bb,
                                                  (short)0, c, false, false);
    }
    // D layout: VGPR r, lanes 0-15 -> M=r (head r), N=lane (n offset). Heads 8..15 padded.
    if (hi == 0) {
        #pragma unroll
        for (int r = 0; r < NH; ++r)
            sbuf[((size_t)(b * NH + r)) * NROWS + n0 + lo] = c[r];
    }
}

// ---------------------------------------------------------------------------
// k3: in-place softmax over n=4096 for each (b,h). 32 blocks x 256 threads.
// ---------------------------------------------------------------------------
__global__ void k3_softmax(float* __restrict__ sbuf) {
    float* s = sbuf + (size_t)blockIdx.x * NROWS;
    int t = threadIdx.x;
    __shared__ float red[256];
    float m = -INFINITY;
    for (int i = t; i < NROWS; i += 256) m = fmaxf(m, s[i]);
    red[t] = m; __syncthreads();
    for (int st = 128; st > 0; st >>= 1) {
        if (t < st) red[t] = fmaxf(red[t], red[t + st]);
        __syncthreads();
    }
    m = red[0]; __syncthreads();
    float sum = 0.f;
    for (int i = t; i < NROWS; i += 256) { float e = expf(s[i] - m); s[i] = e; sum += e; }
    red[t] = sum; __syncthreads();
    for (int st = 128; st > 0; st >>= 1) {
        if (t < st) red[t] += red[t + st];
        __syncthreads();
    }
    float inv = 1.f / red[0];
    for (int i = t; i < NROWS; i += 256) s[i] *= inv;
}

// ---------------------------------------------------------------------------
// k4: z[b,h,:] = sum_n attn[b,h,n] * hn[b,n,:]; y = z @ W_v ; mh = y @ W_mhc
// One block per batch, 128 threads (thread = hid dim).
// ---------------------------------------------------------------------------
__global__ void k4_context(const float* __restrict__ h, const float* __restrict__ attn,
                           const float* __restrict__ Wkv, const float* __restrict__ Wmhc,
                           float* __restrict__ mh) {
    int b = blockIdx.x;
    int t = threadIdx.x;
    __shared__ float aL[NH * 128];
    __shared__ float zL[NH * HID];
    __shared__ float yL[HID];
    float z[NH];
    #pragma unroll
    for (int i = 0; i < NH; ++i) z[i] = 0.f;
    const float* hb = h + (size_t)b * SEQ * HID;
    const float* ab = attn + (size_t)b * NH * NROWS;

    for (int n0 = 0; n0 < NROWS; n0 += 128) {
        __syncthreads();
        #pragma unroll
        for (int j = 0; j < NH; ++j) {
            int idx = t + j * 128;                 // 1024 attn values
            aL[idx] = ab[(idx >> 7) * NROWS + n0 + (idx & 127)];
        }
        __syncthreads();
        for (int i = 0; i < 128; ++i) {
            float hv = hb[(size_t)(n0 + i) * HID + t];
            #pragma unroll
            for (int hh = 0; hh < NH; ++hh) z[hh] += aL[hh * 128 + i] * hv;
        }
    }
    #pragma unroll
    for (int hh = 0; hh < NH; ++hh) zL[hh * HID + t] = z[hh];
    __syncthreads();
    {   // y_flat[t], t = h*16 + u ; V columns are W_kv[:, 128 + t]
        int hh = t >> 4;
        float acc = 0.f;
        for (int d = 0; d < HID; ++d) acc += zL[hh * HID + d] * Wkv[d * 256 + HID + t];
        yL[t] = acc;
    }
    __syncthreads();
    float acc = 0.f;
    for (int i = 0; i < HID; ++i) acc += yL[i] * Wmhc[i * HID + t];
    mh[b * HID + t] = acc;
}

// ---------------------------------------------------------------------------
// k5: y2[b,n] = mh[b,:] . hn[b,n,:]. 64 blocks x 256 threads, float4 row loads.
// ---------------------------------------------------------------------------
__global__ void k5_y2(const float* __restrict__ h, const float* __restrict__ mh,
                      float* __restrict__ y2) {
    int bid = blockIdx.x;
    int b = bid >> 4;
    int n = (bid & 15) * 256 + threadIdx.x;
    __shared__ float mL[HID];
    if (threadIdx.x < HID) mL[threadIdx.x] = mh[b * HID + threadIdx.x];
    __syncthreads();
    const v4f* row = (const v4f*)(h + ((size_t)b * SEQ + n) * HID);
    float acc = 0.f;
    #pragma unroll 8
    for (int i = 0; i < HID / 4; ++i) {
        v4f v = row[i];
        acc += v.x * mL[4 * i] + v.y * mL[4 * i + 1] + v.z * mL[4 * i + 2] + v.w * mL[4 * i + 3];
    }
    y2[b * NROWS + n] = acc;
}

// ---------------------------------------------------------------------------
// k6: out[b, i*4096 + j] = W_lin[0]*y2[b,j] + W_lin[1]*y2[b,i]
// 256 MB streamed with non-temporal stores (output > L2 capacity).
// One block per (b,i) row: 4096 floats = 1024 v4f by 256 threads.
// ---------------------------------------------------------------------------
__global__ void k6_expand(const float* __restrict__ y2, const float* __restrict__ Wlin,
                          float* __restrict__ out) {
    int bid = blockIdx.x;               // 4 * 4096
    int b = bid >> 12;
    int i = bid & 4095;
    float w0 = Wlin[0], w1 = Wlin[1];
    float yi = w1 * y2[b * NROWS + i];
    const v4f* yrow = (const v4f*)(y2 + b * NROWS);
    v4f* orow = (v4f*)(out + ((size_t)(b * NROWS + i)) * NROWS);
    int t = threadIdx.x;
    #pragma unroll
    for (int jj = 0; jj < 4; ++jj) {
        int j4 = jj * 256 + t;          // v4f index 0..1023
        v4f v = yrow[j4];
        v4f r; r.x = v.x * w0 + yi; r.y = v.y * w0 + yi;
               r.z = v.z * w0 + yi; r.w = v.w * w0 + yi;
        __builtin_nontemporal_store(r, &orow[j4]);
    }
}

// ---------------------------------------------------------------------------
extern "C" void kernel_launch(void* const* d_in, const int* in_sizes, int n_in,
                              void* d_out, int out_size, void* d_ws, size_t ws_size,
                              hipStream_t stream) {
    const float* h    = (const float*)d_in[0];   // (4,4097,128)
    const float* Wq   = (const float*)d_in[1];   // (128,128)
    const float* Wkv  = (const float*)d_in[2];   // (128,256)
    const float* Wmhc = (const float*)d_in[3];   // (128,128)
    const float* Wlin = (const float*)d_in[4];   // (2,1)
    float* out = (float*)d_out;                  // (4, 4096*4096, 1)

    // workspace layout (floats); total ~625 KB
    float* ws    = (float*)d_ws;
    float* wfold = ws;                    // 4*16*128      = 8192
    float* sbuf  = wfold + 4 * 16 * HID;  // 4*8*4096      = 131072 (scores -> attn in place)
    float* mh    = sbuf + 4 * NH * NROWS; // 4*128         = 512
    float* y2    = mh + 4 * HID;          // 4*4096        = 16384

    k1_fold   <<<4,         128, 0, stream>>>(h, Wq, Wkv, wfold);
    k2_scores <<<4 * 256,    32, 0, stream>>>(h, wfold, sbuf);
    k3_softmax<<<4 * NH,    256, 0, stream>>>(sbuf);
    k4_context<<<4,         128, 0, stream>>>(h, sbuf, Wkv, Wmhc, mh);
    k5_y2     <<<4 * 16,    256, 0, stream>>>(h, mh, y2);
    k6_expand <<<4 * NROWS, 256, 0, stream>>>(y2, Wlin, out);
}